// GatedWaveMerge_5111011082698
// MI455X (gfx1250) — compile-verified
//
#include <hip/hip_runtime.h>
#include <math.h>
#include <stdint.h>

#define MAX_DEPTH 6

typedef __attribute__((ext_vector_type(2))) float v2f;
typedef __attribute__((ext_vector_type(8))) float v8f;

__device__ __forceinline__ v8f wmma4(v2f a, v2f b, v8f c) {
  // D(16x16,f32) = A(16x4,f32) * B(4x16,f32) + C   -> v_wmma_f32_16x16x4_f32
  return __builtin_amdgcn_wmma_f32_16x16x4_f32(false, a, false, b, (short)0, c, false, false);
}

__device__ __forceinline__ float sigmoidf_(float x) { return 1.0f / (1.0f + __expf(-x)); }

// 4 waves (128 threads) per block; one batch element b per block.
// Heads h=0..15 are the 16 WMMA rows. Waves split the independent WMMA chains
// and the 8 attention head-pairs.
__global__ __launch_bounds__(128) void gated_wave_merge_kernel(
    const float* __restrict__ f_left, const float* __restrict__ f_right,
    const float* __restrict__ wave_freq, const float* __restrict__ wave_damp,
    const float* __restrict__ wave_phase,
    const float* __restrict__ gl_w, const float* __restrict__ gl_b,
    const float* __restrict__ gr_w, const float* __restrict__ gr_b,
    const float* __restrict__ parent_q,
    const float* __restrict__ ln_gamma, const float* __restrict__ ln_beta,
    const float* __restrict__ skip_alpha, const float* __restrict__ skip_w,
    const int* __restrict__ depth_p,
    float* __restrict__ out)
{
  const int tid  = threadIdx.x;
  const int lane = tid & 31;                 // wave32 lane
  const int wave = tid >> 5;                 // 0..3
  const int b    = blockIdx.x;

  // bank rows padded to 68 floats: 272 B = 16-aligned (async B128 dest) and
  // stride%64 = 4 banks -> conflict-free fragment reads.
  __shared__ float bank[16][8][68];          // [h][k(0:4 left | 4:8 rot)][d]
  __shared__ float scratch[4][16][66];       // phase1: gate_in / phase2: per-wave parent rows
  __shared__ float skipBuf[16][66];          // sigmoid(alpha) * left_mean @ skip_w
  __shared__ float sc[4][2][8][9];           // per-wave attn scores/probs (head pair)
  __shared__ float prS[16], piS[16];

  float* const scratchF = &scratch[0][0][0];
  #define GATEIN(m, d)   scratchF[(m) * 130 + (d)]            // 16x128, stride 130
  #define PBUF(w, M, n)  scratchF[(w) * 1056 + (M) * 66 + (n)] // 4 x 16x64, stride 66

  // Pull shared weights toward cache early (global_prefetch_b8)
  __builtin_prefetch(gl_w, 0, 1);
  __builtin_prefetch(gr_w, 0, 1);
  __builtin_prefetch(skip_w, 0, 1);

  const int depth = depth_p[0];
  const int sh = MAX_DEPTH - depth;
  const float scale = (sh > 0) ? (float)(1 << sh) : 1.0f;
  if (tid < 16) {
    int h = tid;
    float x = wave_damp[h];
    float sp = fmaxf(x, 0.0f) + log1pf(__expf(-fabsf(x)));     // softplus
    float alpha_d = sp / scale;
    float omega_d = wave_freq[h] / scale;
    float phi = wave_phase[h] + (float)depth * 0.78539816339744831f; // pi/4
    float dt = scale;
    float decay = __expf(-alpha_d * dt);
    float ang = omega_d * dt + phi;
    prS[h] = decay * __cosf(ang);
    piS[h] = decay * __sinf(ang);
  }
  __syncthreads();

  const float4* fl4 = (const float4*)(f_left  + (size_t)b * 4096);
  const float4* fr4 = (const float4*)(f_right + (size_t)b * 4096);

  // ---- f_left -> bank[:,0:4,:] via async global->LDS B128 (ASYNCcnt path) ----
  for (int i = 0; i < 8; ++i) {
    int idx = i * 128 + tid;                 // float4 index, 0..1023
    int f = idx << 2;
    int h = f >> 8, k = (f >> 6) & 3, d = f & 63;
    uint32_t ldsa = (uint32_t)(uintptr_t)&bank[h][k][d];       // 16B-aligned
    uint64_t ga   = (uint64_t)(uintptr_t)(fl4 + idx);
    asm volatile("global_load_async_to_lds_b128 %0, %1, off"
                 :: "v"(ldsa), "v"(ga) : "memory");
  }
  // ---- f_right rotated -> bank[:,4:8,:] (needs VALU, regular loads) ----
  for (int i = 0; i < 4; ++i) {
    int idx = i * 128 + tid;                 // 0..511 (re-half float4s)
    int f = idx << 2;                        // 0..2047
    int h = f >> 7, k = (f >> 5) & 3, d = f & 31;
    int base4 = (h * 256 + k * 64 + d) >> 2;
    float4 re = fr4[base4];
    float4 im = fr4[base4 + 8];              // +32 floats
    float pr = prS[h], pi_ = piS[h];
    bank[h][4+k][d]    = pr*re.x - pi_*im.x;
    bank[h][4+k][d+1]  = pr*re.y - pi_*im.y;
    bank[h][4+k][d+2]  = pr*re.z - pi_*im.z;
    bank[h][4+k][d+3]  = pr*re.w - pi_*im.w;
    bank[h][4+k][d+32] = pi_*re.x + pr*im.x;
    bank[h][4+k][d+33] = pi_*re.y + pr*im.y;
    bank[h][4+k][d+34] = pi_*re.z + pr*im.z;
    bank[h][4+k][d+35] = pi_*re.w + pr*im.w;
  }
  asm volatile("s_wait_asynccnt 0" ::: "memory");
  __syncthreads();

  // ---- K-means -> gate_in ----
  for (int j = 0; j < 8; ++j) {
    int p = j * 128 + tid;                   // 0..1023
    int m = p >> 6, d = p & 63;
    float sl = bank[m][0][d] + bank[m][1][d] + bank[m][2][d] + bank[m][3][d];
    float sr = bank[m][4][d] + bank[m][5][d] + bank[m][6][d] + bank[m][7][d];
    GATEIN(m, d)      = sl * 0.25f;
    GATEIN(m, 64 + d) = sr * 0.25f;
  }
  __syncthreads();

  const int  row    = lane & 15;
  const bool hiHalf = (lane >= 16);          // 32-bit fragments: hi lanes take K+2,K+3
  const float siga  = sigmoidf_(skip_alpha[0]);

  // ---- 12 independent WMMA chains split across the 4 waves ----
  // c in [0,8): gate matmuls (gate=c>>2, ntile=c&3), 16x128x64
  // c in [8,12): skip matmul ntile=c-8, 16x64x64
  for (int c = wave; c < 12; c += 4) {
    if (c < 8) {
      const int gate  = c >> 2;
      const int nt    = c & 3;
      const float* W    = gate ? gr_w : gl_w;
      const float* bias = gate ? gr_b : gl_b;
      const int kbase   = gate ? 4 : 0;
      int nn = nt * 16 + row;
      v8f acc = {};
      #pragma unroll
      for (int k0 = 0; k0 < 128; k0 += 4) {
        int kk = hiHalf ? k0 + 2 : k0;
        v2f a, bf;
        a.x  = GATEIN(row, kk);     a.y  = GATEIN(row, kk + 1);
        bf.x = W[kk * 64 + nn];     bf.y = W[(kk + 1) * 64 + nn];
        acc = wmma4(a, bf, acc);
      }
      float bv = bias[nn];
      #pragma unroll
      for (int v = 0; v < 8; ++v) {
        int m = hiHalf ? v + 8 : v;          // C layout: hi lanes hold M=v+8
        float g = sigmoidf_(acc[v] + bv);
        bank[m][kbase + 0][nn] *= g;
        bank[m][kbase + 1][nn] *= g;
        bank[m][kbase + 2][nn] *= g;
        bank[m][kbase + 3][nn] *= g;
      }
    } else {
      const int nt = c - 8;
      int nn = nt * 16 + row;
      v8f acc = {};
      #pragma unroll
      for (int k0 = 0; k0 < 64; k0 += 4) {
        int kk = hiHalf ? k0 + 2 : k0;
        v2f a, bf;
        a.x  = GATEIN(row, kk);        a.y  = GATEIN(row, kk + 1);
        bf.x = skip_w[kk * 64 + nn];   bf.y = skip_w[(kk + 1) * 64 + nn];
        acc = wmma4(a, bf, acc);
      }
      #pragma unroll
      for (int v = 0; v < 8; ++v) {
        int m = hiHalf ? v + 8 : v;
        skipBuf[m][nn] = siga * acc[v];
      }
    }
  }
  __syncthreads();

  // ---- attention + parent: 8 head-pairs, 2 per wave ----
  for (int rep = 0; rep < 2; ++rep) {
    int pair = wave * 2 + rep;               // 0..7
    int m0 = pair * 2;
    { // scores: D[q][(msel,k)] = sum_d pq[q&7][d] * bank[m0+msel][k][d]
      v8f acc = {};
      int msel = row >> 3, kb = row & 7;
      #pragma unroll
      for (int k0 = 0; k0 < 64; k0 += 4) {
        int kk = hiHalf ? k0 + 2 : k0;
        v2f a, bf;
        a.x  = parent_q[(row & 7) * 64 + kk];
        a.y  = parent_q[(row & 7) * 64 + kk + 1];
        bf.x = bank[m0 + msel][kb][kk];
        bf.y = bank[m0 + msel][kb][kk + 1];
        acc = wmma4(a, bf, acc);
      }
      if (!hiHalf) {
        #pragma unroll
        for (int v = 0; v < 8; ++v) sc[wave][msel][v][kb] = acc[v] * 0.125f; // 1/sqrt(64)
      }
    }
    __syncthreads();
    if (lane < 16) {                         // softmax over k (8) per (head, q)
      int msel = lane >> 3, q = lane & 7;
      float mx = sc[wave][msel][q][0];
      for (int k = 1; k < 8; ++k) mx = fmaxf(mx, sc[wave][msel][q][k]);
      float e[8], s = 0.0f;
      for (int k = 0; k < 8; ++k) { e[k] = __expf(sc[wave][msel][q][k] - mx); s += e[k]; }
      float inv = 1.0f / s;
      for (int k = 0; k < 8; ++k) sc[wave][msel][q][k] = e[k] * inv;
    }
    __syncthreads();
    // parent: blockdiag(attn_m0, attn_m1)(16x16) @ [bank_m0; bank_m1](16x64)
    for (int nt = 0; nt < 4; ++nt) {
      int nn = nt * 16 + row;
      v8f acc = {};
      #pragma unroll
      for (int k0 = 0; k0 < 16; k0 += 4) {
        int kk = hiHalf ? k0 + 2 : k0;
        float a0, a1;
        if (row < 8) {
          a0 = (kk     < 8) ? sc[wave][0][row][kk]     : 0.0f;
          a1 = (kk + 1 < 8) ? sc[wave][0][row][kk + 1] : 0.0f;
        } else {
          a0 = (kk     >= 8) ? sc[wave][1][row - 8][kk - 8] : 0.0f;
          a1 = (kk + 1 >= 8) ? sc[wave][1][row - 8][kk - 7] : 0.0f;
        }
        v2f a, bf;
        a.x  = a0; a.y = a1;
        bf.x = bank[m0 + (kk >> 3)][kk & 7][nn];
        bf.y = bank[m0 + ((kk + 1) >> 3)][(kk + 1) & 7][nn];
        acc = wmma4(a, bf, acc);
      }
      #pragma unroll
      for (int v = 0; v < 8; ++v) {
        int M = hiHalf ? v + 8 : v;
        PBUF(wave, M, nn) = acc[v];
      }
    }
    __syncthreads();
    // LayerNorm + skip + store (row M: M<8 -> head m0, else m0+1)
    if (lane < 16) {
      int M = lane;
      int m = m0 + (M >> 3);
      int q = M & 7;
      float mu = 0.0f;
      for (int d = 0; d < 64; ++d) mu += PBUF(wave, M, d);
      mu *= (1.0f / 64.0f);
      float var = 0.0f;
      for (int d = 0; d < 64; ++d) { float t = PBUF(wave, M, d) - mu; var += t * t; }
      var *= (1.0f / 64.0f);
      float inv = rsqrtf(var + 1e-5f);
      size_t obase = (((size_t)b * 16 + m) * 8 + q) * 64;
      for (int d = 0; d < 64; ++d) {
        out[obase + d] = (PBUF(wave, M, d) - mu) * inv * ln_gamma[d] + ln_beta[d]
                       + skipBuf[m][d];
      }
    }
    __syncthreads();
  }
  #undef GATEIN
  #undef PBUF
}

extern "C" void kernel_launch(void* const* d_in, const int* in_sizes, int n_in,
                              void* d_out, int out_size, void* d_ws, size_t ws_size,
                              hipStream_t stream) {
  const float* f_left     = (const float*)d_in[0];
  const float* f_right    = (const float*)d_in[1];
  const float* wave_freq  = (const float*)d_in[2];
  const float* wave_damp  = (const float*)d_in[3];
  const float* wave_phase = (const float*)d_in[4];
  const float* gl_w       = (const float*)d_in[5];
  const float* gl_b       = (const float*)d_in[6];
  const float* gr_w       = (const float*)d_in[7];
  const float* gr_b       = (const float*)d_in[8];
  const float* parent_q   = (const float*)d_in[9];
  const float* ln_gamma   = (const float*)d_in[10];
  const float* ln_beta    = (const float*)d_in[11];
  const float* skip_alpha = (const float*)d_in[12];
  const float* skip_w     = (const float*)d_in[13];
  const int*   depth      = (const int*)d_in[14];
  float* out = (float*)d_out;

  int B = in_sizes[0] / (16 * 4 * 64);   // elements per b = H*K*D
  gated_wave_merge_kernel<<<dim3(B), dim3(128), 0, stream>>>(
      f_left, f_right, wave_freq, wave_damp, wave_phase,
      gl_w, gl_b, gr_w, gr_b, parent_q, ln_gamma, ln_beta,
      skip_alpha, skip_w, depth, out);
}